// Single_CNN_GNN_15582141350519
// MI455X (gfx1250) — compile-verified
//
#include <hip/hip_runtime.h>

#define N_NODES 100000
#define N_EDGES 3200000
#define T_DIM 12
#define F_INP 395
#define ROW_STRIDE (T_DIM * F_INP) /* 4740 */
#define SIG_LEN 392
#define C1_POS 96
#define C2_POS 22
#define HIDDEN 20
#define SLOPE 0.2f

#define N_TILES (N_NODES / 16)          /* 6250 */
#define WPB 4                           /* waves per block */
#define NODE_BLOCKS ((N_TILES + WPB - 1) / WPB) /* 1563 */

typedef float v2f __attribute__((ext_vector_type(2)));
typedef float v8f __attribute__((ext_vector_type(8)));

// ---------------- LDS layout (float indices) ----------------
// Shared (block-wide) weights: small matrices only; cnn_lw (112.6 KB) streams
// from L2/WGP$ instead so we can run 4 waves (one per SIMD32) per WGP.
#define W2_OFF 0                        // [96][16]  conv2 weights (B layout)
#define W1_OFF (W2_OFF + 96 * 16)       // [12][16]  conv1 weights, cols 8..15 zero
#define GW_OFF (W1_OFF + 12 * 16)       // [84][32]  gat weights, padded
#define B1_OFF (GW_OFF + 84 * 32)
#define B2_OFF (B1_OFF + 16)
#define LB_OFF (B2_OFF + 16)
#define SH_CONST_END (LB_OFF + 80)      // 4528 floats
// per-wave scratch
#define SIG_STRIDE 399                  // 392 signal / 352 h2 reuse; odd stride
#define H1_STRIDE 769                   // 96*8 conv1 out per node, odd stride
#define FEAT_STRIDE 84
#define SIG_OFF 0
#define H1_OFF (16 * SIG_STRIDE)        // 6384
#define FEAT_OFF H1_OFF                 // alias: H1 dead after conv2
#define HH_OFF (H1_OFF + 16 * FEAT_STRIDE)
#define DUMP_OFF (H1_OFF + 16 * H1_STRIDE)  // 18688: dump slots for invalid cols
#define PW_SIZE (DUMP_OFF + 16)         // 18704 floats
#define LDS_FLOATS (SH_CONST_END + WPB * PW_SIZE)  // 79344 fl = 317,376 B < 320 KB

__device__ __forceinline__ v8f wmma4(v2f a, v2f b, v8f c) {
  // V_WMMA_F32_16X16X4_F32 : D = A(16x4) x B(4x16) + C, exact f32, RNE
  return __builtin_amdgcn_wmma_f32_16x16x4_f32(false, a, false, b, (short)0, c,
                                               false, false);
}

// ============================================================
// Kernel 1: fused per-node pipeline. 4 waves/block, 16 nodes/wave.
// ============================================================
__global__ void node_pipeline(const float* __restrict__ x,
                              const float* __restrict__ w1,
                              const float* __restrict__ b1,
                              const float* __restrict__ w2,
                              const float* __restrict__ b2,
                              const float* __restrict__ lw,
                              const float* __restrict__ lb,
                              const float* __restrict__ gw,
                              const float* __restrict__ att_s,
                              const float* __restrict__ att_d,
                              float* __restrict__ H,
                              float* __restrict__ ASRC,
                              float* __restrict__ ADST) {
  extern __shared__ float sm[];
  const int t = threadIdx.x;

  // ---- stage small weights once per block ([K][N] B-fragment layouts) ----
  for (int i = t; i < 96 * 16; i += 128) {
    int k = i >> 4, n = i & 15;                  // w2: (16,8,12) -> B[k=c*12+t][n=o]
    sm[W2_OFF + i] = w2[n * 96 + k];
  }
  for (int i = t; i < 12 * 16; i += 128) {
    int k = i >> 4, n = i & 15;                  // w1: (8,1,12) -> B[k=tap][n=ch]
    sm[W1_OFF + i] = (n < 8) ? w1[n * 12 + k] : 0.f;
  }
  for (int i = t; i < 84 * 32; i += 128) {
    int k = i >> 5, n = i & 31;
    sm[GW_OFF + i] = (k < 81 && n < 20) ? gw[k * 20 + n] : 0.f;
  }
  if (t < 16) sm[B1_OFF + t] = (t < 8) ? b1[t] : 0.f;
  if (t < 16) sm[B2_OFF + t] = b2[t];
  for (int i = t; i < 80; i += 128) sm[LB_OFF + i] = lb[i];
  __syncthreads();

  const int wave = t >> 5, lane = t & 31;
  const int tile = blockIdx.x * WPB + wave;      // 6252 tiles launched, 6250 valid
  if (tile >= N_TILES) return;                   // wave-uniform; no syncs follow

  const int half = lane >> 4;                    // 0: K0/K1 rows, 1: K2/K3 rows
  const int mrow = lane & 15;                    // node within tile (A rows)
  const int ncol = lane & 15;                    // output column (B/D cols)
  float* PW = sm + SH_CONST_END + wave * PW_SIZE;
  float* SIG = PW + SIG_OFF;
  const int dumpIdx = DUMP_OFF + ncol;           // per-lane garbage slot
  const int node0 = tile * 16;

  // ---- stage the 16 node signals xl[:,3:] (coalesced) ----
  for (int nd = 0; nd < 16; ++nd) {
    const float* row =
        x + (size_t)(node0 + nd) * ROW_STRIDE + (T_DIM - 1) * F_INP + 3;
    for (int j = lane; j < SIG_LEN; j += 32) SIG[nd * SIG_STRIDE + j] = row[j];
  }

  // ---- conv1: per position p, M=16 nodes x N=16(8 ch) x K=12 ----
  {
    const float biasn = sm[B1_OFF + ncol];
    // Loop-invariant branchless store addressing: invalid columns (ncol>=8)
    // point at a dump slot with step 0; valid ones advance 8 floats/position.
    const bool cvalid = (ncol < 8);
    const int stStep = cvalid ? 8 : 0;
    int stBase[8];
#pragma unroll
    for (int r = 0; r < 8; ++r) {
      const int m = r + 8 * half;
      stBase[r] = cvalid ? (H1_OFF + m * H1_STRIDE + ncol) : dumpIdx;
    }
    for (int p = 0; p < C1_POS; ++p) {
      v8f acc;
#pragma unroll
      for (int r = 0; r < 8; ++r) acc[r] = biasn;
#pragma unroll
      for (int kk = 0; kk < 3; ++kk) {
        const int k = kk * 4 + half * 2;
        v2f a, b;
        a[0] = SIG[mrow * SIG_STRIDE + 4 * p + k];
        a[1] = SIG[mrow * SIG_STRIDE + 4 * p + k + 1];
        b[0] = sm[W1_OFF + k * 16 + ncol];
        b[1] = sm[W1_OFF + (k + 1) * 16 + ncol];
        acc = wmma4(a, b, acc);
      }
      const int po = p * stStep;
#pragma unroll
      for (int r = 0; r < 8; ++r) PW[stBase[r] + po] = fmaxf(acc[r], 0.f);
    }
  }

  // ---- conv2: per position p, M=16 x N=16 x K=96 (k = c*12 + tap) ----
  {
    const float biasn = sm[B2_OFF + ncol];
    for (int p = 0; p < C2_POS; ++p) {
      v8f acc;
#pragma unroll
      for (int r = 0; r < 8; ++r) acc[r] = biasn;
#pragma unroll 4
      for (int kk = 0; kk < 24; ++kk) {
        const int k0 = kk * 4 + half * 2;
        const int k1 = k0 + 1;
        const int c0 = k0 / 12, t0 = k0 - 12 * c0;
        const int c1 = k1 / 12, t1 = k1 - 12 * c1;
        v2f a, b;
        a[0] = PW[H1_OFF + mrow * H1_STRIDE + (4 * p + t0) * 8 + c0];
        a[1] = PW[H1_OFF + mrow * H1_STRIDE + (4 * p + t1) * 8 + c1];
        b[0] = sm[W2_OFF + k0 * 16 + ncol];
        b[1] = sm[W2_OFF + k1 * 16 + ncol];
        acc = wmma4(a, b, acc);
      }
      // relu + store flattened (16,22): k = ch*22 + p, reuse SIG region as h2
#pragma unroll
      for (int r = 0; r < 8; ++r) {
        const int m = r + 8 * half;
        SIG[m * SIG_STRIDE + ncol * 22 + p] = fmaxf(acc[r], 0.f);
      }
    }
  }

  // ---- linear: M=16 x K=352 x N=80; B streams from global (L2-resident) ----
#pragma unroll
  for (int nt = 0; nt < 5; ++nt) {
    v8f acc;
    const float biasn = sm[LB_OFF + nt * 16 + ncol];
#pragma unroll
    for (int r = 0; r < 8; ++r) acc[r] = biasn;
#pragma unroll 8
    for (int kk = 0; kk < 88; ++kk) {
      const int k0 = kk * 4 + half * 2;
      v2f a, b;
      a[0] = SIG[mrow * SIG_STRIDE + k0];
      a[1] = SIG[mrow * SIG_STRIDE + k0 + 1];
      b[0] = lw[k0 * 80 + nt * 16 + ncol];
      b[1] = lw[(k0 + 1) * 80 + nt * 16 + ncol];
      acc = wmma4(a, b, acc);
    }
#pragma unroll
    for (int r = 0; r < 8; ++r)   // no activation on the linear output
      PW[FEAT_OFF + (r + 8 * half) * FEAT_STRIDE + nt * 16 + ncol] = acc[r];
  }
  // concat avg_y = x[node, T-1, 0]; pad K 81..83 with zeros
  if (lane < 16) {
    PW[FEAT_OFF + lane * FEAT_STRIDE + 80] =
        x[(size_t)(node0 + lane) * ROW_STRIDE + (T_DIM - 1) * F_INP + 0];
    PW[FEAT_OFF + lane * FEAT_STRIDE + 81] = 0.f;
    PW[FEAT_OFF + lane * FEAT_STRIDE + 82] = 0.f;
    PW[FEAT_OFF + lane * FEAT_STRIDE + 83] = 0.f;
  }

  // ---- GAT matmul: M=16 x K=84 x N=32 (20 valid), no bias here ----
#pragma unroll
  for (int nt = 0; nt < 2; ++nt) {
    v8f acc;
#pragma unroll
    for (int r = 0; r < 8; ++r) acc[r] = 0.f;
#pragma unroll
    for (int kk = 0; kk < 21; ++kk) {
      const int k0 = kk * 4 + half * 2;
      v2f a, b;
      a[0] = PW[FEAT_OFF + mrow * FEAT_STRIDE + k0];
      a[1] = PW[FEAT_OFF + mrow * FEAT_STRIDE + k0 + 1];
      b[0] = sm[GW_OFF + k0 * 32 + ncol + nt * 16];
      b[1] = sm[GW_OFF + (k0 + 1) * 32 + ncol + nt * 16];
      acc = wmma4(a, b, acc);
    }
    const int c = nt * 16 + ncol;
    const bool ok = (c < HIDDEN);
#pragma unroll
    for (int r = 0; r < 8; ++r) {   // branchless: cols >=20 go to dump slot
      const int m = r + 8 * half;
      const int idx = ok ? (HH_OFF + m * HIDDEN + c) : dumpIdx;
      PW[idx] = acc[r];
    }
  }

  // ---- attention logits + spill h ----
  if (lane < 16) {
    float s = 0.f, d0 = 0.f;
#pragma unroll
    for (int c = 0; c < HIDDEN; ++c) {
      const float hv = PW[HH_OFF + lane * HIDDEN + c];
      s += hv * att_s[c];
      d0 += hv * att_d[c];
    }
    ASRC[node0 + lane] = s;
    ADST[node0 + lane] = d0;
  }
  for (int i = lane; i < 16 * HIDDEN; i += 32) {
    const int nd = i / HIDDEN, c = i - nd * HIDDEN;
    H[(size_t)(node0 + nd) * HIDDEN + c] = PW[HH_OFF + nd * HIDDEN + c];
  }
}

// ============================================================
// Edge / aggregation phase (memory + atomic bound)
// ============================================================
__device__ __forceinline__ void atomicMaxF(float* addr, float v) {
  const int iv = __float_as_int(v);
  if (iv >= 0)
    atomicMax((int*)addr, iv);
  else
    atomicMin((unsigned int*)addr, (unsigned int)iv);
}

__global__ void init_aux(float* EMAX, float* ESUM, float* AGG) {
  const int i = blockIdx.x * blockDim.x + threadIdx.x;
  if (i < N_NODES) {
    EMAX[i] = -__builtin_huge_valf();
    ESUM[i] = 0.f;
  }
  if (i < N_NODES * HIDDEN) AGG[i] = -__builtin_huge_valf();
}

__device__ __forceinline__ float edge_logit(const float* ASRC, const float* ADST,
                                            int s, int d) {
  const float v = ASRC[s] + ADST[d];
  return v > 0.f ? v : SLOPE * v;
}

__global__ void edge_max(const long long* __restrict__ ei,
                         const float* __restrict__ ASRC,
                         const float* __restrict__ ADST, float* EMAX) {
  const int i = blockIdx.x * blockDim.x + threadIdx.x;
  if (i >= N_EDGES) return;
  const int s = (int)ei[i], d = (int)ei[N_EDGES + i];
  atomicMaxF(&EMAX[d], edge_logit(ASRC, ADST, s, d));
}

__global__ void edge_sum(const long long* __restrict__ ei,
                         const float* __restrict__ ASRC,
                         const float* __restrict__ ADST,
                         const float* __restrict__ EMAX, float* ESUM) {
  const int i = blockIdx.x * blockDim.x + threadIdx.x;
  if (i >= N_EDGES) return;
  const int s = (int)ei[i], d = (int)ei[N_EDGES + i];
  const float ex = __expf(edge_logit(ASRC, ADST, s, d) - EMAX[d]);
  atomicAdd(&ESUM[d], ex);
}

__global__ void edge_msg(const long long* __restrict__ ei,
                         const float* __restrict__ ASRC,
                         const float* __restrict__ ADST,
                         const float* __restrict__ EMAX,
                         const float* __restrict__ ESUM,
                         const float* __restrict__ H, float* AGG) {
  const int i = blockIdx.x * blockDim.x + threadIdx.x;
  if (i >= N_EDGES) return;
  const int s = (int)ei[i], d = (int)ei[N_EDGES + i];
  const float alpha =
      __expf(edge_logit(ASRC, ADST, s, d) - EMAX[d]) / ESUM[d];
  const float* hs = H + (size_t)s * HIDDEN;
  float* ag = AGG + (size_t)d * HIDDEN;
#pragma unroll
  for (int c = 0; c < HIDDEN; ++c) atomicMaxF(&ag[c], alpha * hs[c]);
}

__global__ void node_out(const float* __restrict__ AGG,
                         const float* __restrict__ gat_b,
                         const float* __restrict__ lin_w,
                         const float* __restrict__ lin_b, float* __restrict__ out) {
  const int n = blockIdx.x * blockDim.x + threadIdx.x;
  if (n >= N_NODES) return;
  float y = 0.f;
#pragma unroll
  for (int c = 0; c < HIDDEN; ++c) {
    float g = AGG[(size_t)n * HIDDEN + c];
    g = (fabsf(g) < __builtin_huge_valf()) ? g : 0.f;  // where(isfinite, g, 0)
    y += fmaxf(g + gat_b[c], 0.f) * lin_w[c];
  }
  out[n] = y + lin_b[0];
}

// ============================================================
extern "C" void kernel_launch(void* const* d_in, const int* in_sizes, int n_in,
                              void* d_out, int out_size, void* d_ws, size_t ws_size,
                              hipStream_t stream) {
  (void)in_sizes; (void)n_in; (void)out_size; (void)ws_size;
  const float* x = (const float*)d_in[0];
  const long long* ei = (const long long*)d_in[1];
  const float* w1 = (const float*)d_in[2];
  const float* b1 = (const float*)d_in[3];
  const float* w2 = (const float*)d_in[4];
  const float* b2 = (const float*)d_in[5];
  const float* lw = (const float*)d_in[6];
  const float* lb = (const float*)d_in[7];
  const float* gw = (const float*)d_in[8];
  const float* ats = (const float*)d_in[9];
  const float* atd = (const float*)d_in[10];
  const float* gb = (const float*)d_in[11];
  const float* lin_w = (const float*)d_in[12];
  const float* lin_b = (const float*)d_in[13];

  float* ws = (float*)d_ws;                 // ~17.6 MB used
  float* H = ws;                            // N*20
  float* ASRC = H + (size_t)N_NODES * HIDDEN;
  float* ADST = ASRC + N_NODES;
  float* EMAX = ADST + N_NODES;
  float* ESUM = EMAX + N_NODES;
  float* AGG = ESUM + N_NODES;              // N*20

  const size_t lds_bytes = (size_t)LDS_FLOATS * sizeof(float);
  node_pipeline<<<NODE_BLOCKS, 128, lds_bytes, stream>>>(
      x, w1, b1, w2, b2, lw, lb, gw, ats, atd, H, ASRC, ADST);
  init_aux<<<(N_NODES * HIDDEN + 255) / 256, 256, 0, stream>>>(EMAX, ESUM, AGG);
  const int eb = (N_EDGES + 255) / 256;
  edge_max<<<eb, 256, 0, stream>>>(ei, ASRC, ADST, EMAX);
  edge_sum<<<eb, 256, 0, stream>>>(ei, ASRC, ADST, EMAX, ESUM);
  edge_msg<<<eb, 256, 0, stream>>>(ei, ASRC, ADST, EMAX, ESUM, H, AGG);
  node_out<<<(N_NODES + 255) / 256, 256, 0, stream>>>(AGG, gb, lin_w, lin_b,
                                                      (float*)d_out);
}